// ScaleInvariantPhotometricLoss_89094801588226
// MI455X (gfx1250) — compile-verified
//
#include <hip/hip_runtime.h>
#include <hip/hip_bf16.h>

// ---------------- problem constants ----------------
#define N_SAMP 32
#define CHN    3
#define IMG_H  384
#define IMG_W  512
#define PLANE  (IMG_H * IMG_W)        // 196608
#define SAMP   (CHN * PLANE)          // 589824
#define NPIX   ((long long)N_SAMP * SAMP) // 18874368

#define C1C 1.0e-4f
#define C2C 9.0e-4f

// ---------------- tiling ----------------
#define TW 32
#define THH 8
#define HALO_W 34
#define HALO_H 10
#define TILES_X (IMG_W / TW)              // 16
#define TILES_Y (IMG_H / THH)             // 48
#define TILES_PER_PLANE (TILES_X * TILES_Y) // 768
#define NPLANES (N_SAMP * CHN)            // 96
#define NTILES  (NPLANES * TILES_PER_PLANE) // 73728

#define STAT_CHUNKS 36
#define STAT_BLOCKS (2 * N_SAMP * STAT_CHUNKS) // 2304
#define CHUNK_F4 4096                          // 16384 floats per chunk
#define SSIM_BLOCKS 4608                       // 73728 / 4608 = 16 tiles per block

typedef float v2f __attribute__((ext_vector_type(2)));
typedef float v8f __attribute__((ext_vector_type(8)));

// address-space-qualified pointer types for the async global->LDS builtin
typedef __attribute__((address_space(1))) int* as1_i32p;
typedef __attribute__((address_space(3))) int* as3_i32p;

// ---------------- wave reductions ----------------
__device__ __forceinline__ float wave_sum(float v) {
  v += __shfl_xor(v, 1, 32);
  v += __shfl_xor(v, 2, 32);
  v += __shfl_xor(v, 4, 32);
  v += __shfl_xor(v, 8, 32);
  v += __shfl_xor(v, 16, 32);
  return v;
}

// Wave-wide add using the f32 WMMA: A = ones(16x4), B holds one value per lane
// (second VGPR zero). Sum over the whole 16x16 C tile equals 16 * sum(B),
// independent of the exact A/B lane layout. Requires EXEC all ones.
__device__ __forceinline__ float wave_sum_wmma(float v) {
#if __has_builtin(__builtin_amdgcn_wmma_f32_16x16x4_f32)
  v2f a; a[0] = 1.0f; a[1] = 1.0f;
  v2f b; b[0] = v;    b[1] = 0.0f;
  v8f c = {};
  c = __builtin_amdgcn_wmma_f32_16x16x4_f32(false, a, false, b, (short)0, c,
                                            false, false);
  float s = c[0] + c[1] + c[2] + c[3] + c[4] + c[5] + c[6] + c[7];
  s += __shfl_xor(s, 1, 32);
  s += __shfl_xor(s, 2, 32);
  s += __shfl_xor(s, 4, 32);
  s += __shfl_xor(s, 8, 32);
  s += __shfl_xor(s, 16, 32);
  return s * (1.0f / 16.0f);
#else
  return wave_sum(v);
#endif
}

// ---------------- async global->LDS staging ----------------
#if __has_builtin(__builtin_amdgcn_global_load_async_to_lds_b32)
#define USE_ASYNC_LDS 1
#endif

__device__ __forceinline__ void wait_async_lds() {
#if defined(USE_ASYNC_LDS)
#if __has_builtin(__builtin_amdgcn_s_wait_asynccnt)
  __builtin_amdgcn_s_wait_asynccnt(0);
#else
  asm volatile("s_wait_asynccnt 0x0" ::: "memory");
#endif
#endif
}

__device__ __forceinline__ void stage_pair_to_lds(const float* g1,
                                                  const float* g2, long off,
                                                  float* l1, float* l2) {
#if defined(USE_ASYNC_LDS)
  __builtin_amdgcn_global_load_async_to_lds_b32((as1_i32p)(g1 + off),
                                                (as3_i32p)l1, 0, 0);
  __builtin_amdgcn_global_load_async_to_lds_b32((as1_i32p)(g2 + off),
                                                (as3_i32p)l2, 0, 0);
#else
  *l1 = g1[off];
  *l2 = g2[off];
#endif
}

// ---------------- kernel 1: per-chunk sum / sumsq ----------------
__global__ __launch_bounds__(256) void ssim_stats_partial(
    const float* __restrict__ x0, const float* __restrict__ x1,
    float* __restrict__ part) {
  const int b = blockIdx.x;
  const int tid = threadIdx.x;
  const int ti = b / (N_SAMP * STAT_CHUNKS);
  const int rem = b % (N_SAMP * STAT_CHUNKS);
  const int s = rem / STAT_CHUNKS;
  const int chunk = rem % STAT_CHUNKS;
  const float* src = (ti == 0 ? x0 : x1) + (size_t)s * SAMP +
                     (size_t)chunk * (SAMP / STAT_CHUNKS);
  const float4* v = (const float4*)src;
  float s1 = 0.0f, s2 = 0.0f;
  for (int i = tid; i < CHUNK_F4; i += 256) {
    float4 q = v[i];
    s1 += q.x + q.y + q.z + q.w;
    s2 += q.x * q.x + q.y * q.y + q.z * q.z + q.w * q.w;
  }
  s1 = wave_sum(s1);
  s2 = wave_sum(s2);
  __shared__ float ls[8][2];
  const int lane = tid & 31, wid = tid >> 5;
  if (lane == 0) { ls[wid][0] = s1; ls[wid][1] = s2; }
  __syncthreads();
  if (tid == 0) {
    float a = 0.0f, c = 0.0f;
    for (int i = 0; i < 8; ++i) { a += ls[i][0]; c += ls[i][1]; }
    part[b * 2 + 0] = a;
    part[b * 2 + 1] = c;
  }
}

// ---------------- kernel 2: finalize mu / 1/(std+eps) ----------------
__global__ __launch_bounds__(32) void ssim_stats_final(
    const float* __restrict__ part, float* __restrict__ stats) {
  const int b = blockIdx.x;  // 0..63 = ti*32 + sample
  const int tid = threadIdx.x;
  float s1 = 0.0f, s2 = 0.0f;
  for (int i = tid; i < STAT_CHUNKS; i += 32) {
    s1 += part[(b * STAT_CHUNKS + i) * 2 + 0];
    s2 += part[(b * STAT_CHUNKS + i) * 2 + 1];
  }
  s1 = wave_sum(s1);
  s2 = wave_sum(s2);
  if (tid == 0) {
    const float n = (float)SAMP;
    const float mu = s1 / n;
    float var = (s2 - n * mu * mu) / (n - 1.0f);
    var = fmaxf(var, 0.0f);
    const float sd = sqrtf(var);
    stats[b * 2 + 0] = mu;
    stats[b * 2 + 1] = 1.0f / (sd + 1e-8f);
  }
}

// ---------------- kernel 3: fused SSIM over tiles ----------------
__global__ __launch_bounds__(256) void ssim_map_partial(
    const float* __restrict__ rec, const float* __restrict__ tgt,
    const float* __restrict__ stats, float* __restrict__ part) {
  __shared__ float raw1[HALO_H][36];
  __shared__ float raw2[HALO_H][36];
  __shared__ float h_r [HALO_H][TW];
  __shared__ float h_t [HALO_H][TW];
  __shared__ float h_r2[HALO_H][TW];
  __shared__ float h_t2[HALO_H][TW];
  __shared__ float h_rt[HALO_H][TW];
  __shared__ float wsum[8];

  const int tid = threadIdx.x;
  float acc = 0.0f;

  for (int t = blockIdx.x; t < NTILES; t += SSIM_BLOCKS) {
    const int p  = t / TILES_PER_PLANE;
    const int tp = t % TILES_PER_PLANE;
    const int ty = tp / TILES_X;
    const int tx = tp % TILES_X;
    const int smp = p / CHN;
    const float mu1  = stats[smp * 2 + 0];
    const float isd1 = stats[smp * 2 + 1];
    const float mu2  = stats[(N_SAMP + smp) * 2 + 0];
    const float isd2 = stats[(N_SAMP + smp) * 2 + 1];
    const float* g1 = rec + (size_t)p * PLANE;
    const float* g2 = tgt + (size_t)p * PLANE;
    const int y0 = ty * THH - 1;
    const int x0 = tx * TW - 1;
    // block-uniform: tile + halo fully inside the image?
    const bool interior =
        (tx > 0) & (tx < TILES_X - 1) & (ty > 0) & (ty < TILES_Y - 1);

    // ---- stage 1: stage raw halo tile into LDS (async where available) ----
    if (interior) {
      // full-EXEC async issue, no per-lane predication
      for (int idx = tid; idx < HALO_W * HALO_H; idx += 256) {
        const int row = idx / HALO_W, col = idx % HALO_W;
        const long off = (long)(y0 + row) * IMG_W + (x0 + col);
        stage_pair_to_lds(g1, g2, off, &raw1[row][col], &raw2[row][col]);
      }
    } else {
      for (int idx = tid; idx < HALO_W * HALO_H; idx += 256) {
        const int row = idx / HALO_W, col = idx % HALO_W;
        const int gy = y0 + row, gx = x0 + col;
        const bool valid = (gy >= 0) & (gy < IMG_H) & (gx >= 0) & (gx < IMG_W);
        if (valid) {
          const long off = (long)gy * IMG_W + gx;
          stage_pair_to_lds(g1, g2, off, &raw1[row][col], &raw2[row][col]);
        }
#if !defined(USE_ASYNC_LDS)
        else {
          raw1[row][col] = 0.0f;
          raw2[row][col] = 0.0f;
        }
#endif
      }
    }
    wait_async_lds();
    __syncthreads();

    // ---- stage 2: horizontal 3-tap sums of r, t, r^2, t^2, r*t ----
    if (interior) {
      for (int idx = tid; idx < TW * HALO_H; idx += 256) {
        const int row = idx / TW, x = idx % TW;
        const float a0 = raw1[row][x], a1 = raw1[row][x + 1],
                    a2 = raw1[row][x + 2];
        const float b0 = raw2[row][x], b1 = raw2[row][x + 1],
                    b2 = raw2[row][x + 2];
        const float r0 = (a0 - mu1) * isd1, r1 = (a1 - mu1) * isd1,
                    r2 = (a2 - mu1) * isd1;
        const float t0 = (b0 - mu2) * isd2, t1 = (b1 - mu2) * isd2,
                    t2 = (b2 - mu2) * isd2;
        h_r [row][x] = r0 + r1 + r2;
        h_t [row][x] = t0 + t1 + t2;
        h_r2[row][x] = r0 * r0 + r1 * r1 + r2 * r2;
        h_t2[row][x] = t0 * t0 + t1 * t1 + t2 * t2;
        h_rt[row][x] = r0 * t0 + r1 * t1 + r2 * t2;
      }
    } else {
      for (int idx = tid; idx < TW * HALO_H; idx += 256) {
        const int row = idx / TW, x = idx % TW;
        const int gy = y0 + row;
        const bool yv = (gy >= 0) & (gy < IMG_H);
        // unconditional LDS reads (always in-bounds in the padded LDS tile);
        // zero-padding of the *normalized* image applied via branchless selects
        float a[3], bv[3];
#pragma unroll
        for (int j = 0; j < 3; ++j) {
          a[j] = raw1[row][x + j];
          bv[j] = raw2[row][x + j];
        }
        float sr = 0.0f, st = 0.0f, sr2 = 0.0f, st2 = 0.0f, srt = 0.0f;
#pragma unroll
        for (int j = 0; j < 3; ++j) {
          const int gx = x0 + x + j;
          const bool v = yv & (gx >= 0) & (gx < IMG_W);
          const float rj = v ? (a[j] - mu1) * isd1 : 0.0f;
          const float tj = v ? (bv[j] - mu2) * isd2 : 0.0f;
          sr += rj; st += tj;
          sr2 += rj * rj; st2 += tj * tj; srt += rj * tj;
        }
        h_r [row][x] = sr;  h_t [row][x] = st;
        h_r2[row][x] = sr2; h_t2[row][x] = st2;
        h_rt[row][x] = srt;
      }
    }
    __syncthreads();

    // ---- stage 3: vertical 3-tap sums + SSIM formula (one pixel/thread) ----
    {
      const int x = tid % TW, y = tid / TW;
      const float S1  = h_r [y][x] + h_r [y + 1][x] + h_r [y + 2][x];
      const float S2  = h_t [y][x] + h_t [y + 1][x] + h_t [y + 2][x];
      const float Sr2 = h_r2[y][x] + h_r2[y + 1][x] + h_r2[y + 2][x];
      const float St2 = h_t2[y][x] + h_t2[y + 1][x] + h_t2[y + 2][x];
      const float Srt = h_rt[y][x] + h_rt[y + 1][x] + h_rt[y + 2][x];
      const float inv9 = 1.0f / 9.0f;
      const float m1 = S1 * inv9, m2 = S2 * inv9;
      const float m1s = m1 * m1, m2s = m2 * m2, m12 = m1 * m2;
      const float sig1 = Sr2 * inv9 - m1s;
      const float sig2 = St2 * inv9 - m2s;
      const float s12  = Srt * inv9 - m12;
      const float num = (2.0f * m12 + C1C) * (2.0f * s12 + C2C);
      const float den = (m1s + m2s + C1C) * (sig1 + sig2 + C2C);
      const float m = num / den;
      acc += fminf(fmaxf((1.0f - m) * 0.5f, 0.0f), 1.0f);
    }
    __syncthreads();  // protect LDS before next tile's stage 1
  }

  // ---- block reduction: WMMA wave-wide add, then 8 wave partials ----
  const float s = wave_sum_wmma(acc);
  const int lane = tid & 31, wid = tid >> 5;
  if (lane == 0) wsum[wid] = s;
  __syncthreads();
  if (tid == 0) {
    float tot = 0.0f;
    for (int i = 0; i < 8; ++i) tot += wsum[i];
    part[blockIdx.x] = tot;
  }
}

// ---------------- kernel 4: final mean ----------------
__global__ __launch_bounds__(256) void ssim_final(
    const float* __restrict__ part, float* __restrict__ out) {
  const int tid = threadIdx.x;
  float acc = 0.0f;
  for (int i = tid; i < SSIM_BLOCKS; i += 256) acc += part[i];
  const float s = wave_sum_wmma(acc);
  __shared__ float wsum[8];
  if ((tid & 31) == 0) wsum[tid >> 5] = s;
  __syncthreads();
  if (tid == 0) {
    float tot = 0.0f;
    for (int i = 0; i < 8; ++i) tot += wsum[i];
    out[0] = tot * (1.0f / (float)NPIX);
  }
}

// ---------------- launch ----------------
extern "C" void kernel_launch(void* const* d_in, const int* in_sizes, int n_in,
                              void* d_out, int out_size, void* d_ws,
                              size_t ws_size, hipStream_t stream) {
  const float* rec = (const float*)d_in[0];
  const float* tgt = (const float*)d_in[1];
  float* out = (float*)d_out;
  float* ws = (float*)d_ws;
  float* statsPart = ws;                 // 2304*2 = 4608 floats
  float* stats     = ws + 4608;          // 64*2   = 128 floats
  float* ssimPart  = ws + 4608 + 128;    // 4608 floats
  (void)in_sizes; (void)n_in; (void)out_size; (void)ws_size;

  ssim_stats_partial<<<STAT_BLOCKS, 256, 0, stream>>>(rec, tgt, statsPart);
  ssim_stats_final<<<64, 32, 0, stream>>>(statsPart, stats);
  ssim_map_partial<<<SSIM_BLOCKS, 256, 0, stream>>>(rec, tgt, stats, ssimPart);
  ssim_final<<<1, 256, 0, stream>>>(ssimPart, out);
}